// AlignToZWignerD_7232724927076
// MI455X (gfx1250) — compile-verified
//
#include <hip/hip_runtime.h>

// ---------------------------------------------------------------------------
// AlignToZ Wigner-D (l = 0..4, DIM = 25) for 200k points.
// Store-bandwidth-bound: 500 MB output -> ~21.5 us floor at 23.3 TB/s.
// Strategy: 1 wave32 per WG, 1 point per lane, recurrence fully unrolled in
// registers, tile staged in LDS in the exact output layout, streamed out with
// CDNA5 async LDS->global b128 stores (coalesced 512B per issue).
// ---------------------------------------------------------------------------

#define PTS    32               // points per workgroup (== wave32 lanes)
#define ROWDW  625              // dwords per point (25x25)
#define TILEDW (PTS * ROWDW)    // 20000 dwords = 80 KB LDS tile

#if defined(__has_builtin)
#  if __has_builtin(__builtin_amdgcn_global_store_async_from_lds_b128)
#    define HAVE_ASYNC128 1
#  endif
#  if __has_builtin(__builtin_amdgcn_global_store_async_from_lds_b32)
#    define HAVE_ASYNC32 1
#  endif
#  if __has_builtin(__builtin_amdgcn_s_wait_asynccnt)
#    define HAVE_WAITASYNC 1
#  endif
#endif

// Exact pointee types the async builtins expect (from hipcc diagnostics):
//   b128: AS1/AS3 pointer to "int __attribute__((vector_size(16)))"
//   b32 : AS1/AS3 pointer to "int"
typedef int v4i_t __attribute__((vector_size(16)));
typedef __attribute__((address_space(1))) v4i_t gv4_t;  // global int4
typedef __attribute__((address_space(3))) v4i_t lv4_t;  // lds    int4
typedef __attribute__((address_space(1))) int   gi_t;   // global int
typedef __attribute__((address_space(3))) int   li_t;   // lds    int

// P(i,l,a,b) term of the Ivanic-Ruedenberg recurrence.
// dp = D^{l-1} ((2l-1)^2, row-major), r1 = 3x3 (rows m=-1,0,1).
template <int L>
__device__ __forceinline__ float wP(int i, int a, int b, const float* dp,
                                    const float* r1) {
  constexpr int LP = L - 1;
  constexpr int NP = 2 * LP + 1;
  const int row = i + 1;
  if (b == L)
    return r1[row * 3 + 2] * dp[(a + LP) * NP + 2 * LP] -
           r1[row * 3 + 0] * dp[(a + LP) * NP + 0];
  if (b == -L)
    return r1[row * 3 + 2] * dp[(a + LP) * NP + 0] +
           r1[row * 3 + 0] * dp[(a + LP) * NP + 2 * LP];
  return r1[row * 3 + 1] * dp[(a + LP) * NP + (b + LP)];
}

// Build D^L from D^{L-1} and R1.  Fully unrolled: all coefficients constant
// fold (sqrt of integer ratios) and all array indices are compile-time.
template <int L>
__device__ __forceinline__ void wigner_step(const float* dp, const float* r1,
                                            float* dl) {
  constexpr int NN = 2 * L + 1;
#pragma unroll
  for (int m = -L; m <= L; ++m) {
#pragma unroll
    for (int mp = -L; mp <= L; ++mp) {
      const int am = (m < 0) ? -m : m;
      const int amp = (mp < 0) ? -mp : mp;
      const double denom = (amp == L) ? (double)((2 * L) * (2 * L - 1))
                                      : (double)((L + mp) * (L - mp));
      const bool m0 = (m == 0);
      const double u = __builtin_sqrt((double)((L + m) * (L - m)) / denom);
      const double vmag =
          0.5 * __builtin_sqrt((m0 ? 2.0 : 1.0) *
                               (double)((L + am - 1) * (L + am)) / denom);
      const double v = m0 ? -vmag : vmag;  // *(1 - 2*d0)
      const int wn = (((L - am - 1) > 0 ? (L - am - 1) : 0) * (L - am));
      const double w = m0 ? 0.0 : (-0.5 * __builtin_sqrt((double)wn / denom));

      float val = 0.0f;
      if (u != 0.0) val += (float)u * wP<L>(0, m, mp, dp, r1);

      if (m == 0) {
        val += (float)v * wP<L>(1, 1, mp, dp, r1);
        val += (float)v * wP<L>(-1, -1, mp, dp, r1);
      } else if (m > 0) {
        const double s = (m == 1) ? 1.4142135623730951 : 1.0;
        val += (float)(v * s) * wP<L>(1, m - 1, mp, dp, r1);
        if (m != 1) val += (float)(-v) * wP<L>(-1, -m + 1, mp, dp, r1);
      } else {  // m < 0
        if (m != -1) val += (float)v * wP<L>(1, m + 1, mp, dp, r1);
        const double s = (m == -1) ? 1.4142135623730951 : 1.0;
        val += (float)(v * s) * wP<L>(-1, -m - 1, mp, dp, r1);
      }

      if (w != 0.0) {
        if (m > 0) {
          val += (float)w * wP<L>(1, m + 1, mp, dp, r1);
          val += (float)w * wP<L>(-1, -m - 1, mp, dp, r1);
        } else {
          val += (float)w * wP<L>(1, m - 1, mp, dp, r1);
          val += (float)(-w) * wP<L>(-1, -m + 1, mp, dp, r1);
        }
      }
      dl[(m + L) * NN + (mp + L)] = val;
    }
  }
}

__global__ __launch_bounds__(PTS) void wignerd_kernel(
    const float* __restrict__ xyz, float* __restrict__ out, int n) {
  __shared__ __align__(16) float lds[TILEDW];
  const int lane = threadIdx.x;                       // 0..31
  const int g = blockIdx.x * PTS + lane;              // global point

  // ---- zero the staged tile (linear b128, conflict-free, 16B aligned) ----
  {
    float4* l4 = reinterpret_cast<float4*>(lds);
    const float4 z4 = make_float4(0.f, 0.f, 0.f, 0.f);
    for (int i = 0; i < 156; ++i) l4[i * 32 + lane] = z4;   // 19968 dwords
    lds[19968 + lane] = 0.0f;                               // tail 32 dwords
  }

  // ---- per-lane recurrence ----
  if (g < n) {
    const float x = xyz[3 * g + 0];
    const float y = xyz[3 * g + 1];
    const float zc = xyz[3 * g + 2];
    const float nrm = __builtin_sqrtf(x * x + y * y + zc * zc);
    const float inv = 1.0f / fmaxf(nrm, 1e-14f);
    const float vx = x * inv, vy = y * inv, vz = zc * inv;
    const float ct = fminf(fmaxf(vz, -1.0f), 1.0f);
    const float st = __builtin_sqrtf(fmaxf(1.0f - ct * ct, 0.0f));
    const float rxy = __builtin_sqrtf(vx * vx + vy * vy);
    float cp, sp;
    if (rxy > 0.0f) { cp = vx / rxy; sp = vy / rxy; }
    else            { cp = 1.0f;     sp = 0.0f;     }

    // R1 = perm(Ry(-theta)Rz(-phi)) in real-SH order m=(-1,0,1)
    float r1[9];
    r1[0] = cp;       r1[1] = 0.0f; r1[2] = -sp;
    r1[3] = st * sp;  r1[4] = ct;   r1[5] = st * cp;
    r1[6] = ct * sp;  r1[7] = -st;  r1[8] = ct * cp;

    float* rowp = lds + lane * ROWDW;  // this lane's 625-dword image
    rowp[0] = 1.0f;  // l=0
#pragma unroll
    for (int r = 0; r < 3; ++r)
#pragma unroll
      for (int c = 0; c < 3; ++c) rowp[26 + 25 * r + c] = r1[3 * r + c];

    float d2[25];
    wigner_step<2>(r1, r1, d2);
#pragma unroll
    for (int r = 0; r < 5; ++r)
#pragma unroll
      for (int c = 0; c < 5; ++c) rowp[104 + 25 * r + c] = d2[5 * r + c];

    float d3[49];
    wigner_step<3>(d2, r1, d3);
#pragma unroll
    for (int r = 0; r < 7; ++r)
#pragma unroll
      for (int c = 0; c < 7; ++c) rowp[234 + 25 * r + c] = d3[7 * r + c];

    float d4[81];
    wigner_step<4>(d3, r1, d4);
#pragma unroll
    for (int r = 0; r < 9; ++r)
#pragma unroll
      for (int c = 0; c < 9; ++c) rowp[416 + 25 * r + c] = d4[9 * r + c];
  }

  __syncthreads();  // all DS writes visible before the streaming copy

  // ---- stream tile to global, fully coalesced ----
  const long long base = (long long)blockIdx.x * TILEDW;   // dwords
  const long long total = (long long)n * ROWDW;
  float* og = out + base;

  if (base + TILEDW <= total) {
#if defined(HAVE_ASYNC128) && defined(HAVE_ASYNC32)
#pragma unroll 4
    for (int i = 0; i < 156; ++i) {
      const int dw = (i * 32 + lane) * 4;  // 16B-aligned, lanes contiguous
      __builtin_amdgcn_global_store_async_from_lds_b128(
          (gv4_t*)(og + dw), (lv4_t*)(lds + dw), 0, 0);
    }
    __builtin_amdgcn_global_store_async_from_lds_b32(
        (gi_t*)(og + 19968 + lane), (li_t*)(lds + 19968 + lane), 0, 0);
#if defined(HAVE_WAITASYNC)
    __builtin_amdgcn_s_wait_asynccnt(0);
#endif
#else
    float4* o4 = reinterpret_cast<float4*>(og);
    const float4* s4 = reinterpret_cast<const float4*>(lds);
#pragma unroll 4
    for (int i = 0; i < 156; ++i) {
      const int idx = i * 32 + lane;
      o4[idx] = s4[idx];
    }
    og[19968 + lane] = lds[19968 + lane];
#endif
  } else {
    // generic tail (never taken for n = 200000)
    for (int i = lane; i < TILEDW; i += PTS) {
      const long long dw = base + i;
      if (dw < total) out[dw] = lds[i];
    }
  }
}

extern "C" void kernel_launch(void* const* d_in, const int* in_sizes, int n_in,
                              void* d_out, int out_size, void* d_ws,
                              size_t ws_size, hipStream_t stream) {
  (void)n_in; (void)out_size; (void)d_ws; (void)ws_size;
  const float* xyz = (const float*)d_in[0];
  float* out = (float*)d_out;
  const int n = in_sizes[0] / 3;
  const int blocks = (n + PTS - 1) / PTS;
  wignerd_kernel<<<blocks, PTS, 0, stream>>>(xyz, out, n);
}